// Layer_70858370449689
// MI455X (gfx1250) — compile-verified
//
#include <hip/hip_runtime.h>

// ---------------------------------------------------------------------------
// GraphSAGE layer for MI455X (gfx1250, wave32, WMMA)
//   msg = feat[src] * ew ; agg = segsum(msg, dst) ; h = agg/max(deg,1)
//   out = [h | feat] @ W^T + b        (W: [256, 512] row-major)
// ---------------------------------------------------------------------------

typedef __attribute__((ext_vector_type(2))) float v2f;
typedef __attribute__((ext_vector_type(4))) float f32x4;
typedef __attribute__((ext_vector_type(8))) float v8f;

#define N_NODES 10000
#define N_EDGES 320000
#define D_IN    256
#define D_OUT   256
#define K_DIM   512   // 2 * D_IN

// ---------------------------------------------------------------- zero ws --
__global__ void zero_f32(float* __restrict__ p, int n) {
    int i = blockIdx.x * blockDim.x + threadIdx.x;
    if (i < n) p[i] = 0.0f;
}

// ----------------------------------------------------- W [256,512] -> Wt ---
// Wt[k*256 + n] = W[n*512 + k]   (k-major so B-fragment loads are coalesced)
__global__ void transpose_w(const float* __restrict__ W, float* __restrict__ Wt) {
    int i = blockIdx.x * blockDim.x + threadIdx.x;
    if (i < D_OUT * K_DIM) {
        int n = i / K_DIM;
        int k = i - n * K_DIM;
        Wt[k * D_OUT + n] = W[i];
    }
}

// ----------------------------------------------------------- edge scatter --
// one wave32 per edge; lane covers 8 of 256 features via two float4 gathers
__global__ void __launch_bounds__(256)
edge_scatter(const float* __restrict__ feat,
             const float* __restrict__ ew,
             const long long* __restrict__ src,
             const long long* __restrict__ dst,
             float* __restrict__ agg,
             float* __restrict__ deg) {
    int gid  = blockIdx.x * blockDim.x + threadIdx.x;
    int edge = gid >> 5;
    int lane = threadIdx.x & 31;
    if (edge >= N_EDGES) return;

    long long s = src[edge];
    long long d = dst[edge];
    float     w = ew[edge];

    const float* fs = feat + (size_t)s * D_IN;
    float*       ad = agg  + (size_t)d * D_IN;

#pragma unroll
    for (int half = 0; half < 2; ++half) {
        int base = lane * 4 + half * 128;
        f32x4 v = *(const f32x4*)(fs + base);
        atomicAdd(ad + base + 0, v.x * w);
        atomicAdd(ad + base + 1, v.y * w);
        atomicAdd(ad + base + 2, v.z * w);
        atomicAdd(ad + base + 3, v.w * w);
    }
    if (lane == 0) atomicAdd(deg + d, 1.0f);
}

// --------------------------------------------------------- mean normalize --
__global__ void normalize_agg(float* __restrict__ agg, const float* __restrict__ deg) {
    int i = blockIdx.x * blockDim.x + threadIdx.x;
    if (i < N_NODES * D_IN) {
        int n = i >> 8;                 // i / 256
        agg[i] = agg[i] / fmaxf(deg[n], 1.0f);
    }
}

// ------------------------------------------------------------- WMMA GEMM ---
// out[N,256] = [hneigh | feat] @ Wt + b, accumulated with v_wmma_f32_16x16x4_f32.
// Block = 128 threads (4 waves). Block b owns rows [b*16, b*16+16).
// Wave w owns output cols [w*64, w*64+64) as four 16x16 WMMA tiles.
//
// Fragment layouts (CDNA5 ISA 7.12.2, 32-bit, wave32):
//   A 16x4 : lane L -> row m=L%16 ; v0: k=kb+(L/16)*2, v1: k=kb+(L/16)*2+1
//   B 4x16 : lane L -> col n=L%16 ; v0: k=kb+(L/16)*2, v1: k=kb+(L/16)*2+1
//   C 16x16: vgpr r, lane L -> row = r + (L/16)*8, col = L%16
__global__ void __launch_bounds__(128)
sage_gemm_wmma(const float* __restrict__ hneigh,  // [N, 256] (agg, normalized)
               const float* __restrict__ feat,    // [N, 256]
               const float* __restrict__ Wt,      // [512, 256] k-major
               const float* __restrict__ bias,    // [256]
               float* __restrict__ out) {         // [N, 256]
    const int lane      = threadIdx.x & 31;
    const int wave      = threadIdx.x >> 5;        // 0..3
    const int row_base  = blockIdx.x * 16;         // 625 blocks cover 10000 rows
    const int ncol_base = wave * 64;

    const int m     = lane & 15;
    const int khalf = (lane >> 4) * 2;             // 0 or 2

    v8f acc[4] = {v8f{}, v8f{}, v8f{}, v8f{}};

    for (int k = 0; k < K_DIM; k += 4) {
        // A fragment from the virtual concat [hneigh | feat] (uniform select,
        // k crosses 256 only on a 4-aligned boundary so the pick is wave-uniform)
        const float* Arow = (k < D_IN)
            ? (hneigh + (size_t)(row_base + m) * D_IN + (k + khalf))
            : (feat   + (size_t)(row_base + m) * D_IN + (k - D_IN + khalf));
        v2f a = *(const v2f*)Arow;                  // 8B aligned (k,khalf even)

        const float* Bk0 = Wt + (size_t)(k + khalf)     * D_OUT;
        const float* Bk1 = Wt + (size_t)(k + khalf + 1) * D_OUT;

#pragma unroll
        for (int t = 0; t < 4; ++t) {
            int n = ncol_base + t * 16 + (lane & 15);
            v2f b;
            b.x = Bk0[n];
            b.y = Bk1[n];
            acc[t] = __builtin_amdgcn_wmma_f32_16x16x4_f32(
                /*neg_a=*/false, a, /*neg_b=*/false, b,
                /*c_mod=*/(short)0, acc[t], /*reuse_a=*/false, /*reuse_b=*/false);
        }
    }

    const int mrow = (lane >> 4) * 8;
#pragma unroll
    for (int t = 0; t < 4; ++t) {
        int   n  = ncol_base + t * 16 + (lane & 15);
        float bb = bias[n];
#pragma unroll
        for (int r = 0; r < 8; ++r) {
            out[(size_t)(row_base + mrow + r) * D_OUT + n] = acc[t][r] + bb;
        }
    }
}

// ---------------------------------------------------------------------------
extern "C" void kernel_launch(void* const* d_in, const int* in_sizes, int n_in,
                              void* d_out, int out_size, void* d_ws, size_t ws_size,
                              hipStream_t stream) {
    const float*     feat = (const float*)d_in[0];      // [10000, 256]
    const float*     ew   = (const float*)d_in[1];      // [320000, 1]
    const float*     W    = (const float*)d_in[2];      // [256, 512]
    const float*     bias = (const float*)d_in[3];      // [256]
    const long long* src  = (const long long*)d_in[4];  // [320000] int64
    const long long* dst  = (const long long*)d_in[5];  // [320000] int64
    float*           out  = (float*)d_out;              // [10000, 256]

    // workspace layout (floats): agg[N*256] | deg[N] | Wt[512*256]
    float* agg = (float*)d_ws;
    float* deg = agg + (size_t)N_NODES * D_IN;
    float* Wt  = deg + N_NODES;

    // 1) zero accumulators (agg + deg are contiguous)
    {
        int n = N_NODES * D_IN + N_NODES;
        zero_f32<<<(n + 255) / 256, 256, 0, stream>>>(agg, n);
    }
    // 2) transpose W -> Wt (k-major)
    {
        int n = D_OUT * K_DIM;
        transpose_w<<<(n + 255) / 256, 256, 0, stream>>>(W, Wt);
    }
    // 3) edge scatter: one wave per edge
    {
        int waves  = N_EDGES;
        int blocks = (waves * 32 + 255) / 256;          // 40000
        edge_scatter<<<blocks, 256, 0, stream>>>(feat, ew, src, dst, agg, deg);
    }
    // 4) mean normalize
    {
        int n = N_NODES * D_IN;
        normalize_agg<<<(n + 255) / 256, 256, 0, stream>>>(agg, deg);
    }
    // 5) WMMA GEMM: 625 blocks x 128 threads (4 waves), wave = 16x64 tile
    sage_gemm_wmma<<<N_NODES / 16, 128, 0, stream>>>(agg, feat, Wt, bias, out);
}